// RNN_64424509440217
// MI455X (gfx1250) — compile-verified
//
#include <hip/hip_runtime.h>

// Problem constants (from reference)
#define BATCH  128
#define TSTEPS 1024
#define ISZ    100
#define HSZ    200
#define NROWS  (BATCH * TSTEPS)          // 131072 rows for the big GEMM
#define COLT   13                        // ceil(200/16) column tiles

typedef float v2f __attribute__((ext_vector_type(2)));
typedef float v8f __attribute__((ext_vector_type(8)));

// D = A(16x4,f32) * B(4x16,f32) + C(16x16,f32) -- true fp32 WMMA on gfx1250
__device__ __forceinline__ v8f wmma_f32x4(v2f a, v2f b, v8f c) {
  return __builtin_amdgcn_wmma_f32_16x16x4_f32(
      /*neg_a=*/false, a, /*neg_b=*/false, b,
      /*c_mod=*/(short)0, c, /*reuse_a=*/false, /*reuse_b=*/false);
}

// ---------------------------------------------------------------------------
// Kernel 1: x_proj[r][h] = input[r][:] . W_ih[h][:] + b_ih[h] + b_hh[h]
// written directly into the hiddens region of d_out (consumed in-place by scan).
// One wave per 16x16 output tile; K loop 100 = 25 x 4.
// ---------------------------------------------------------------------------
__global__ void __launch_bounds__(256)
xproj_kernel(const float* __restrict__ input,
             const float* __restrict__ W_ih,
             const float* __restrict__ b_ih,
             const float* __restrict__ b_hh,
             float* __restrict__ hiddens) {
  const int lane = threadIdx.x & 31;
  const int wave = threadIdx.x >> 5;
  const int tile = blockIdx.x * 8 + wave;           // 106496 tiles total
  const int rowTile = tile / COLT;
  const int colTile = tile - rowTile * COLT;
  const long row0 = (long)rowTile * 16;
  const int  col0 = colTile * 16;
  const int  n    = lane & 15;
  const int  half = lane >> 4;
  const int  colA = col0 + n;
  const int  colC = colA < (HSZ - 1) ? colA : (HSZ - 1);   // clamped (tail tile)

  const float* Arow = input + (row0 + n) * ISZ + 2 * half;   // A: m = lane%16
  const float* Brow = W_ih + (long)colC * ISZ + 2 * half;    // B[k][n] = W_ih[n][k]

  v8f acc = {};
#pragma unroll 5
  for (int k0 = 0; k0 < ISZ; k0 += 4) {
    v2f a = *(const v2f*)(Arow + k0);
    v2f b = *(const v2f*)(Brow + k0);
    acc = wmma_f32x4(a, b, acc);
  }

  const float bias = b_ih[colC] + b_hh[colC];
  if (colA < HSZ) {
#pragma unroll
    for (int j = 0; j < 8; ++j) {
      const long r = row0 + j + 8 * half;           // D: M = j + 8*half
      hiddens[r * HSZ + colA] = acc[j] + bias;
    }
  }
}

// ---------------------------------------------------------------------------
// Kernel 2: persistent single-workgroup recurrent scan.
//   h_t = relu(x_proj_t + h_{t-1} @ W_hh^T), hiddens[b][t][:] updated in place.
// 26 waves: wave w -> column tile (w>>1), row tiles [(w&1)*4 .. +3].
// Each wave keeps 4 tiles of f32 accumulators; B fragments of W_hh are reused
// across the 4 row tiles. Workgroup barrier + workgroup-scope fence per step.
// ---------------------------------------------------------------------------
__global__ void __launch_bounds__(832)
scan_kernel(const float* __restrict__ W_hh,
            float* __restrict__ hiddens) {
  const int lane = threadIdx.x & 31;
  const int wave = threadIdx.x >> 5;                // 0..25
  const int colTile = wave >> 1;                    // 0..12
  const int rtBase  = (wave & 1) * 4;               // 0 or 4 (batch-row tiles)
  const int col0 = colTile * 16;
  const int n    = lane & 15;
  const int half = lane >> 4;
  const int colA = col0 + n;
  const int colC = colA < (HSZ - 1) ? colA : (HSZ - 1);
  const bool valid = colA < HSZ;

  const float* Bbase = W_hh + (long)colC * HSZ + 2 * half;   // B[k][n]=W_hh[n][k]

  for (int t = 0; t < TSTEPS; ++t) {
    v8f acc[4];
    // Load x_proj tile (this wave's own output location at time t).
#pragma unroll
    for (int rt = 0; rt < 4; ++rt) {
#pragma unroll
      for (int j = 0; j < 8; ++j) {
        const long brow = (long)(rtBase + rt) * 16 + j + 8 * half;
        acc[rt][j] = valid ? hiddens[(brow * TSTEPS + t) * HSZ + colA] : 0.0f;
      }
    }

    if (t > 0) {
      // A base pointers: rows b, columns k of h_{t-1}
      const float* Abase[4];
#pragma unroll
      for (int rt = 0; rt < 4; ++rt) {
        const long mrow = (long)(rtBase + rt) * 16 + n;      // A: m = lane%16
        Abase[rt] = hiddens + (mrow * TSTEPS + (t - 1)) * HSZ + 2 * half;
      }
      for (int k0 = 0; k0 < HSZ; k0 += 4) {                  // 50 iterations
        const v2f b = *(const v2f*)(Bbase + k0);
#pragma unroll
        for (int rt = 0; rt < 4; ++rt) {
          const v2f a = *(const v2f*)(Abase[rt] + k0);
          acc[rt] = wmma_f32x4(a, b, acc[rt]);
        }
      }
    }

    // relu + store h_t
    if (valid) {
#pragma unroll
      for (int rt = 0; rt < 4; ++rt) {
#pragma unroll
        for (int j = 0; j < 8; ++j) {
          const long brow = (long)(rtBase + rt) * 16 + j + 8 * half;
          const float v = acc[rt][j];
          hiddens[(brow * TSTEPS + t) * HSZ + colA] = v > 0.0f ? v : 0.0f;
        }
      }
    }

    __threadfence_block();   // workgroup-scope release of the global stores
    __syncthreads();         // all waves see h_t before step t+1
  }
}

// ---------------------------------------------------------------------------
// Kernel 3: outputs[r] = hiddens[r][:] . W_out + b_out   (r = b*T + t)
// ---------------------------------------------------------------------------
__global__ void __launch_bounds__(256)
out_kernel(const float* __restrict__ hiddens,
           const float* __restrict__ W_out,
           const float* __restrict__ b_out,
           float* __restrict__ out) {
  const long r = (long)blockIdx.x * blockDim.x + threadIdx.x;
  if (r >= (long)NROWS) return;
  const float4* hp = (const float4*)(hiddens + r * HSZ);
  const float4* wp = (const float4*)(W_out);
  float s = 0.0f;
#pragma unroll 10
  for (int i = 0; i < HSZ / 4; ++i) {
    const float4 h = hp[i];
    const float4 w = wp[i];
    s += h.x * w.x + h.y * w.y + h.z * w.z + h.w * w.w;
  }
  out[r] = s + b_out[0];
}

extern "C" void kernel_launch(void* const* d_in, const int* in_sizes, int n_in,
                              void* d_out, int out_size, void* d_ws, size_t ws_size,
                              hipStream_t stream) {
  const float* input = (const float*)d_in[0];   // [B,T,100]
  const float* W_ih  = (const float*)d_in[1];   // [200,100]
  const float* W_hh  = (const float*)d_in[2];   // [200,200]
  const float* b_ih  = (const float*)d_in[3];   // [200]
  const float* b_hh  = (const float*)d_in[4];   // [200]
  const float* W_out = (const float*)d_in[5];   // [1,200]
  const float* b_out = (const float*)d_in[6];   // [1]

  float* outputs = (float*)d_out;                    // [B,T,1]  = 131072 floats
  float* hiddens = outputs + (long)NROWS;            // [B,T,200]

  // 1) x_proj + biases -> hiddens region (8192*13 tiles, 8 waves/block)
  xproj_kernel<<<(NROWS / 16) * COLT / 8, 256, 0, stream>>>(
      input, W_ih, b_ih, b_hh, hiddens);

  // 2) persistent single-WGP recurrent scan over 1024 steps
  scan_kernel<<<1, 832, 0, stream>>>(W_hh, hiddens);

  // 3) output head
  out_kernel<<<(NROWS + 255) / 256, 256, 0, stream>>>(
      hiddens, W_out, b_out, outputs);
}